// LocallyConnected2dV2_4423816315098
// MI455X (gfx1250) — compile-verified
//
#include <hip/hip_runtime.h>

typedef float v2f __attribute__((ext_vector_type(2)));
typedef float v8f __attribute__((ext_vector_type(8)));

#define IN_H  128
#define IN_W  128
#define HP    132
#define WP    132
#define NKH   25
#define NKW   25
#define NK    625   // NKH*NKW
#define KH    10
#define TAPW  88    // padded j-extent per di per 16-window tile (85 -> 88, mult of 4)

// One wave computes a 16-batch x 16-window output tile with f32 WMMA.
// Tile id: bt (batch tile, 64) x r (window row, 25) x ct (col tile, 2: c0 = 0 or 9).
__global__ __launch_bounds__(256) void lc2d_wmma_f32(
    const float* __restrict__ x,     // [1024,128,128]
    const float* __restrict__ W,     // [17424,625] row-major, structured-sparse dense
    const float* __restrict__ bias,  // [25,25]
    float* __restrict__ out)         // [1024,25,25]
{
    const int lane   = threadIdx.x & 31;
    const int waveId = threadIdx.x >> 5;
    const int tile   = blockIdx.x * 8 + waveId;   // 0..3199

    const int ct = tile & 1;
    const int r  = (tile >> 1) % NKH;
    const int bt = tile / (2 * NKH);

    const int hi = lane >> 4;    // 0 or 1
    const int n  = lane & 15;    // M-index for A, N-index for B/D

    const int c0 = ct * 9;            // window-col tile start (0 or 9)
    const int k0 = r * NKW + c0;      // W column base
    const int j0 = 5 * c0;            // padded-x column base (0 or 45)
    const int b0 = bt * 16;           // batch base
    const int r5 = 5 * r;             // padded-x row base

    const float* xb = x + (size_t)(b0 + n) * (IN_H * IN_W);

    v8f acc0 = {};
    v8f acc1 = {};

    for (int di = 0; di < KH; ++di) {
        const int row = r5 + di;              // padded row, 0..129
        const int ir  = row - 2;              // input row, -2..127 (never >127)
        if (ir < 0) continue;                 // wave-uniform (only r==0, di<2): A slice is 0

        const float* xrow = xb + ir * IN_W;
        // Per-lane A base: interior taps live at constant offsets 4*jc from here.
        const float* pA = xrow + (j0 - 2 + 2 * hi);
        // Per-lane B base: taps at constant offsets 4*NK*kk from here.
        const float* wrow = W + (size_t)(row * WP + j0) * NK + (k0 + n);

        #pragma unroll
        for (int jc = 0; jc < TAPW; jc += 4) {
            float a0, a1;
            if (jc == 0 || jc == TAPW - 4) {
                // Edge steps: left pad (j0==0, hi==0) or right edge (j0==45),
                // where an unclamped load could read outside the x buffer.
                const int ic0 = j0 + jc + 2 * hi - 2;
                const int ic1 = ic0 + 1;
                const int cc0 = ic0 < 0 ? 0 : (ic0 > IN_W - 1 ? IN_W - 1 : ic0);
                const int cc1 = ic1 < 0 ? 0 : (ic1 > IN_W - 1 ? IN_W - 1 : ic1);
                a0 = (ic0 == cc0) ? xrow[cc0] : 0.0f;   // v_cndmask, EXEC untouched
                a1 = (ic1 == cc1) ? xrow[cc1] : 0.0f;
            } else {
                // Interior: immediate-offset loads, no masking needed.
                a0 = pA[jc];
                a1 = pA[jc + 1];
            }

            const int kk = jc + 2 * hi;
            const float bb0 = wrow[(size_t)kk * NK];
            const float bb1 = wrow[(size_t)(kk + 1) * NK];

            v2f av = {a0, a1};
            v2f bv = {bb0, bb1};
            if (di & 1)
                acc1 = __builtin_amdgcn_wmma_f32_16x16x4_f32(
                    false, av, false, bv, (short)0, acc1, false, false);
            else
                acc0 = __builtin_amdgcn_wmma_f32_16x16x4_f32(
                    false, av, false, bv, (short)0, acc0, false, false);
        }
    }

    v8f d = acc0 + acc1;
    const float bv = bias[k0 + n];   // bias[r, c0+n]

    #pragma unroll
    for (int i = 0; i < 8; ++i) {
        const int m = i + 8 * hi;    // batch offset within tile (D layout)
        out[(size_t)(b0 + m) * NK + (k0 + n)] = d[i] + bv;
    }
}

extern "C" void kernel_launch(void* const* d_in, const int* in_sizes, int n_in,
                              void* d_out, int out_size, void* d_ws, size_t ws_size,
                              hipStream_t stream) {
    const float* x    = (const float*)d_in[0];   // [1024,128,128]
    const float* W    = (const float*)d_in[1];   // [17424,625]
    const float* bias = (const float*)d_in[2];   // [25,25]
    float* out = (float*)d_out;                  // [1024,25,25]

    // 3200 wave-tiles = 400 blocks x 8 waves
    dim3 grid(400), block(256);
    hipLaunchKernelGGL(lc2d_wmma_f32, grid, block, 0, stream, x, W, bias, out);
}